// Kmeans_Loss_21114059227768
// MI455X (gfx1250) — compile-verified
//
#include <hip/hip_runtime.h>
#include <math.h>

// ---------------- problem constants ----------------
#define N_SAMPLES 262144
#define DIM 256
#define NCLUS 196
#define CPAD 208          // 13 * 16
#define TILES 13
#define INV_T 2.0f        // 1 / TEMPERATURE (0.5)
#define EPS 1e-12f

// ---------------- segment-sum tiling ----------------
#define DCHUNK 64
#define NCHUNK (DIM / DCHUNK)                         // 4
#define SAMPLE_BLOCKS 64
#define SAMPLES_PER_BLOCK (N_SAMPLES / SAMPLE_BLOCKS) // 4096

// ---------------- workspace layout (float offsets) ----------------
#define OFF_QSUM 0
#define OFF_KSUM (OFF_QSUM + NCLUS * DIM)     // 50176
#define OFF_CNT  (OFF_KSUM + NCLUS * DIM)     // 100352
#define OFF_QC   (OFF_CNT + 256)              // 100608 (aligned, padded centers)
#define OFF_KC   (OFF_QC + CPAD * DIM)        // 153856
#define OFF_DQ   (OFF_KC + CPAD * DIM)        // 207104
#define OFF_DK   (OFF_DQ + CPAD * CPAD)       // 250368
#define OFF_ROWLOSS (OFF_DK + CPAD)           // 250576
#define WS_FLOATS (OFF_ROWLOSS + CPAD)        // ~250784 floats ~= 1 MB

typedef __attribute__((ext_vector_type(2))) float v2f;
typedef __attribute__((ext_vector_type(8))) float v8f;

// ---------------- kernels ----------------

__global__ void zero_ws_kernel(float* __restrict__ ws) {
    const unsigned i = blockIdx.x * blockDim.x + threadIdx.x;
    if (i < WS_FLOATS) ws[i] = 0.0f;
}

// Segment sums with LDS privatization.
// grid = (SAMPLE_BLOCKS, NCHUNK, 2), block = 256.
__global__ void accumulate_kernel(const float* __restrict__ im_q,
                                  const float* __restrict__ im_k,
                                  const int* __restrict__ labels,
                                  float* __restrict__ ws) {
    __shared__ float lsum[NCLUS * DCHUNK];   // ~49 KB
    __shared__ float lcnt[NCLUS];
    const int t     = threadIdx.x;
    const int chunk = blockIdx.y;
    const int z     = blockIdx.z;
    const bool do_cnt = (z == 0) && (chunk == 0);
    const float* __restrict__ in = (z == 0) ? im_q : im_k;
    float* __restrict__ gsum = ws + ((z == 0) ? OFF_QSUM : OFF_KSUM);

    for (int i = t; i < NCLUS * DCHUNK; i += 256) lsum[i] = 0.0f;
    if (t < NCLUS) lcnt[t] = 0.0f;
    __syncthreads();

    const int d  = t & (DCHUNK - 1);
    const int sl = t >> 6;                        // 4 samples in flight
    const int base = blockIdx.x * SAMPLES_PER_BLOCK;
    const int coff = chunk * DCHUNK + d;
    for (int it = 0; it < SAMPLES_PER_BLOCK; it += 4) {
        const int s   = base + it + sl;
        const int lbl = labels[s];
        const float x = in[(size_t)s * DIM + coff];
        atomicAdd(&lsum[lbl * DCHUNK + d], x);    // ds_add_f32
        if (do_cnt && d == 0) atomicAdd(&lcnt[lbl], 1.0f);
    }
    __syncthreads();

    for (int i = t; i < NCLUS * DCHUNK; i += 256) {
        const int c  = i >> 6;
        const int dl = i & (DCHUNK - 1);
        atomicAdd(&gsum[c * DIM + chunk * DCHUNK + dl], lsum[i]);
    }
    if (do_cnt && t < NCLUS) atomicAdd(&ws[OFF_CNT + t], lcnt[t]);
}

// centers = (sums / max(count,eps)) then L2-normalize.  grid=(196,2), block=256.
__global__ void finalize_centers_kernel(float* __restrict__ ws) {
    __shared__ float red[256];
    const int c = blockIdx.x;
    const int z = blockIdx.y;
    const int t = threadIdx.x;
    const float* __restrict__ sums = ws + ((z == 0) ? OFF_QSUM : OFF_KSUM);
    float* __restrict__ ctr = ws + ((z == 0) ? OFF_QC : OFF_KC);
    const float cnt = ws[OFF_CNT + c];
    const float v   = sums[c * DIM + t] / fmaxf(cnt, EPS);
    red[t] = v * v;
    __syncthreads();
    for (int s = 128; s > 0; s >>= 1) {
        if (t < s) red[t] += red[t + s];
        __syncthreads();
    }
    const float nrm = fmaxf(sqrtf(red[0]), EPS);
    ctr[c * DIM + t] = v / nrm;
}

// d_k[c] = dot(q_centers[c], k_centers[c]) / T.  grid=196, block=256.
__global__ void dk_kernel(float* __restrict__ ws) {
    __shared__ float red[256];
    const int c = blockIdx.x;
    const int t = threadIdx.x;
    red[t] = ws[OFF_QC + c * DIM + t] * ws[OFF_KC + c * DIM + t];
    __syncthreads();
    for (int s = 128; s > 0; s >>= 1) {
        if (t < s) red[t] += red[t + s];
        __syncthreads();
    }
    if (t == 0) ws[OFF_DK + c] = red[0] * INV_T;
}

// d_q = (Qc @ Qc^T) / T via V_WMMA_F32_16X16X4_F32.
// grid=(13,13), block=32 (one full wave, EXEC all-ones).
__global__ void wmma_gemm_kernel(float* __restrict__ ws) {
    const float* __restrict__ Q = ws + OFF_QC;
    float* __restrict__ dq = ws + OFF_DQ;
    const int lane = threadIdx.x;
    const int m    = lane & 15;
    const int half = lane >> 4;
    // A: 16x4 f32 -> 2 VGPRs: VGPR0 = K (0|2), VGPR1 = K (1|3) by lane half.
    const float* __restrict__ pa = Q + (size_t)(blockIdx.x * 16 + m) * DIM + 2 * half;
    // B: 4x16 f32, row-per-lane-stripe, symmetric layout; B[k][n] = Q[n][k].
    const float* __restrict__ pb = Q + (size_t)(blockIdx.y * 16 + m) * DIM + 2 * half;
    v8f acc = {};
    for (int k = 0; k < DIM; k += 4) {
        const v2f a = *(const v2f*)(pa + k);
        const v2f b = *(const v2f*)(pb + k);
        acc = __builtin_amdgcn_wmma_f32_16x16x4_f32(
            /*neg_a=*/false, a, /*neg_b=*/false, b,
            /*c_mod=*/(short)0, acc, /*reuse_a=*/false, /*reuse_b=*/false);
    }
    // D layout: VGPR r -> M = r + 8*half, N = lane&15
#pragma unroll
    for (int r = 0; r < 8; ++r) {
        const int mo = blockIdx.x * 16 + r + 8 * half;
        const int no = blockIdx.y * 16 + m;
        dq[mo * CPAD + no] = acc[r] * INV_T;
    }
}

// Per-row: diag <- d_k, mask zero-count columns to -10, logsumexp.  grid=196, block=256.
__global__ void row_loss_kernel(float* __restrict__ ws) {
    __shared__ float red[256];
    const int i = blockIdx.x;
    const int t = threadIdx.x;
    const float* __restrict__ dq  = ws + OFF_DQ;
    const float* __restrict__ dk  = ws + OFF_DK;
    const float* __restrict__ cnt = ws + OFF_CNT;
    const bool zi = (cnt[i] == 0.0f);
    float v = -INFINITY;
    if (t < NCLUS) {
        float val = (t == i) ? dk[i] : dq[i * CPAD + t];
        if (cnt[t] == 0.0f) val = -10.0f;
        v = val;
    }
    red[t] = v;
    __syncthreads();
    for (int s = 128; s > 0; s >>= 1) {
        if (t < s) red[t] = fmaxf(red[t], red[t + s]);
        __syncthreads();
    }
    const float M = red[0];
    __syncthreads();
    red[t] = (t < NCLUS) ? expf(v - M) : 0.0f;
    __syncthreads();
    for (int s = 128; s > 0; s >>= 1) {
        if (t < s) red[t] += red[t + s];
        __syncthreads();
    }
    if (t == 0) {
        const float pos = zi ? -10.0f : dk[i];
        ws[OFF_ROWLOSS + i] = zi ? 0.0f : (-pos + M + logf(red[0]));
    }
}

// Deterministic serial final reduction -> scalar loss.
__global__ void final_kernel(const float* __restrict__ ws, float* __restrict__ out) {
    float s = 0.0f, nz = 0.0f;
    for (int c = 0; c < NCLUS; ++c) {
        s += ws[OFF_ROWLOSS + c];
        if (ws[OFF_CNT + c] == 0.0f) nz += 1.0f;
    }
    out[0] = s / ((float)NCLUS - nz);
}

// ---------------- launcher ----------------
extern "C" void kernel_launch(void* const* d_in, const int* in_sizes, int n_in,
                              void* d_out, int out_size, void* d_ws, size_t ws_size,
                              hipStream_t stream) {
    const float* im_q   = (const float*)d_in[0];
    const float* im_k   = (const float*)d_in[1];
    const int*   labels = (const int*)d_in[2];
    float* ws  = (float*)d_ws;
    float* out = (float*)d_out;

    zero_ws_kernel<<<(WS_FLOATS + 255) / 256, 256, 0, stream>>>(ws);

    dim3 gacc(SAMPLE_BLOCKS, NCHUNK, 2);
    accumulate_kernel<<<gacc, 256, 0, stream>>>(im_q, im_k, labels, ws);

    finalize_centers_kernel<<<dim3(NCLUS, 2), 256, 0, stream>>>(ws);
    dk_kernel<<<NCLUS, 256, 0, stream>>>(ws);
    wmma_gemm_kernel<<<dim3(TILES, TILES), 32, 0, stream>>>(ws);
    row_loss_kernel<<<NCLUS, 256, 0, stream>>>(ws);
    final_kernel<<<1, 1, 0, stream>>>(ws, out);
}